// EdgeMultiHeadAttention_9414568312873
// MI455X (gfx1250) — compile-verified
//
#include <hip/hip_runtime.h>
#include <hip/hip_bf16.h>

// ---------------------------------------------------------------------------
// EdgeMultiHeadAttention for MI455X (gfx1250, wave32, WMMA f32 16x16x4)
//   B=2, S=2000, HIDDEN=1024, HEADS=16, d=64, KNN=20, EDGE_DICT=64
// Pipeline:
//   1) qh/kh/vh = x @ W^T + b     (WMMA f32 GEMM, 32x32 per wave, LDS-tiled)
//   2) table[64][2] = edge_emb @ mask_emb^T  (bias collapses to 128 values)
//   3) flash attention per (b,h,16-row q-tile), K/V staged via
//      GLOBAL_LOAD_ASYNC_TO_LDS_B128 (ASYNCcnt path)
//   4) out = ao @ Wo^T + bo       (same GEMM kernel)
// ---------------------------------------------------------------------------

typedef __attribute__((ext_vector_type(2))) float v2f;
typedef __attribute__((ext_vector_type(8))) float v8f;

#define HID   1024
#define NHEAD 16
#define DH    64
#define SEQ   2000
#define NB    2
#define MROWS (NB * SEQ)             // 4000
#define SEG   ((size_t)MROWS * HID)  // floats per [4000,1024] tensor

#define USE_ASYNC_LDS 1

// LDS byte offset of a __shared__ object (addrspace(3) view).
__device__ __forceinline__ unsigned lds_off(const void* p) {
    return (unsigned)(uintptr_t)(__attribute__((address_space(3))) const char*)p;
}

#define WMMA_F32(A, B, C) \
    __builtin_amdgcn_wmma_f32_16x16x4_f32(false, (A), false, (B), (short)0, (C), false, false)

// ------------------------------ projection GEMM ----------------------------
// O[m,n] = sum_k X[m,k] * W[n,k] + bias[n];  N = K = 1024, M = 4000.
// Block: 256 threads = 8 waves (2M x 4N), block tile 64 x 128, K-chunk 64.
// Each wave: 2x2 grid of 16x16 WMMA C tiles (32x32) -> A/B fragments reused 2x.
__global__ __launch_bounds__(256) void proj_gemm_kernel(
    const float* __restrict__ X, const float* __restrict__ W,
    const float* __restrict__ bias, float* __restrict__ O, int M)
{
    __shared__ float As[64][68];    // stride 68: conflict-free fragment reads
    __shared__ float Ws[128][68];

    const int tid  = threadIdx.x;
    const int lane = tid & 31;
    const int wave = tid >> 5;         // 0..7
    const int half = lane >> 4;        // 0/1
    const int l16  = lane & 15;
    const int klo  = half * 2;         // f32 WMMA: lanes16-31 hold K+2,K+3

    const int m0 = blockIdx.y * 64;
    const int n0 = blockIdx.x * 128;
    const int wm = (wave >> 2) * 32;   // 0 or 32
    const int wn = (wave & 3) * 32;    // 0,32,64,96

    v8f c00 = {}, c01 = {}, c10 = {}, c11 = {};

    // staging assignments
    const int ar = tid >> 2;           // 0..63   (A row)
    const int ac = (tid & 3) * 16;     // 0,16,32,48
    const int wr = tid >> 1;           // 0..127  (W row)
    const int wc = (tid & 1) * 32;     // 0,32
    const int arow = (m0 + ar < M) ? (m0 + ar) : (M - 1);  // clamp OOB rows

    for (int k0 = 0; k0 < HID; k0 += 64) {
        // --- stage A tile 64x64 : 16 floats / thread ---
        {
            const float4* src = reinterpret_cast<const float4*>(
                X + (size_t)arow * HID + k0 + ac);
            float4 t0 = src[0], t1 = src[1], t2 = src[2], t3 = src[3];
            *reinterpret_cast<float4*>(&As[ar][ac])      = t0;
            *reinterpret_cast<float4*>(&As[ar][ac + 4])  = t1;
            *reinterpret_cast<float4*>(&As[ar][ac + 8])  = t2;
            *reinterpret_cast<float4*>(&As[ar][ac + 12]) = t3;
        }
        // --- stage W tile 128x64 : 32 floats / thread ---
        {
            const float4* src = reinterpret_cast<const float4*>(
                W + (size_t)(n0 + wr) * HID + k0 + wc);
            float4 t0 = src[0], t1 = src[1], t2 = src[2], t3 = src[3];
            float4 t4 = src[4], t5 = src[5], t6 = src[6], t7 = src[7];
            *reinterpret_cast<float4*>(&Ws[wr][wc])      = t0;
            *reinterpret_cast<float4*>(&Ws[wr][wc + 4])  = t1;
            *reinterpret_cast<float4*>(&Ws[wr][wc + 8])  = t2;
            *reinterpret_cast<float4*>(&Ws[wr][wc + 12]) = t3;
            *reinterpret_cast<float4*>(&Ws[wr][wc + 16]) = t4;
            *reinterpret_cast<float4*>(&Ws[wr][wc + 20]) = t5;
            *reinterpret_cast<float4*>(&Ws[wr][wc + 24]) = t6;
            *reinterpret_cast<float4*>(&Ws[wr][wc + 28]) = t7;
        }
        __syncthreads();

#pragma unroll
        for (int kk = 0; kk < 64; kk += 4) {
            v2f a0, a1, b0, b1;
            a0.x = As[wm + l16][kk + klo];      a0.y = As[wm + l16][kk + klo + 1];
            a1.x = As[wm + 16 + l16][kk + klo]; a1.y = As[wm + 16 + l16][kk + klo + 1];
            b0.x = Ws[wn + l16][kk + klo];      b0.y = Ws[wn + l16][kk + klo + 1];
            b1.x = Ws[wn + 16 + l16][kk + klo]; b1.y = Ws[wn + 16 + l16][kk + klo + 1];
            c00 = WMMA_F32(a0, b0, c00);
            c01 = WMMA_F32(a0, b1, c01);
            c10 = WMMA_F32(a1, b0, c10);
            c11 = WMMA_F32(a1, b1, c11);
        }
        __syncthreads();
    }

    const float bv0 = bias[n0 + wn + l16];
    const float bv1 = bias[n0 + wn + 16 + l16];
#pragma unroll
    for (int i = 0; i < 8; ++i) {
        const int r0 = m0 + wm + i + 8 * half;   // C row = vgpr + 8*half
        const int r1 = r0 + 16;
        if (r0 < M) {
            O[(size_t)r0 * HID + n0 + wn + l16]      = c00[i] + bv0;
            O[(size_t)r0 * HID + n0 + wn + 16 + l16] = c01[i] + bv1;
        }
        if (r1 < M) {
            O[(size_t)r1 * HID + n0 + wn + l16]      = c10[i] + bv0;
            O[(size_t)r1 * HID + n0 + wn + 16 + l16] = c11[i] + bv1;
        }
    }
}

// ------------------------------ bias table ---------------------------------
// table[e][c] = dot(edge_emb[e,:], mask_emb[c,:]), e<64, c<2
__global__ __launch_bounds__(256) void bias_table_kernel(
    const float* __restrict__ edge_emb, const float* __restrict__ mask_emb,
    float* __restrict__ tbl)
{
    __shared__ float red[2][8];
    const int e = blockIdx.x;
    const int t = threadIdx.x;
    float s0 = 0.f, s1 = 0.f;
    for (int k = t; k < HID; k += 256) {
        const float ev = edge_emb[(size_t)e * HID + k];
        s0 += ev * mask_emb[k];
        s1 += ev * mask_emb[HID + k];
    }
#pragma unroll
    for (int off = 16; off > 0; off >>= 1) {
        s0 += __shfl_xor(s0, off, 32);
        s1 += __shfl_xor(s1, off, 32);
    }
    if ((t & 31) == 0) { red[0][t >> 5] = s0; red[1][t >> 5] = s1; }
    __syncthreads();
    if (t == 0) {
        float a = 0.f, b = 0.f;
#pragma unroll
        for (int w = 0; w < 8; ++w) { a += red[0][w]; b += red[1][w]; }
        tbl[e * 2 + 0] = a;
        tbl[e * 2 + 1] = b;
    }
}

// ------------------------------ flash attention ----------------------------
// One wave per (b, h, 16-row q tile). Online softmax over 125 k-tiles of 16.
__global__ __launch_bounds__(32) void attn_flash_kernel(
    const float* __restrict__ qh, const float* __restrict__ kh,
    const float* __restrict__ vh, const int* __restrict__ cmask,
    const int* __restrict__ degv, const float* __restrict__ tbl,
    float* __restrict__ ao)
{
    __shared__ float Ks[16][68];
    __shared__ float Vs[16][68];
    __shared__ float Ps[16][17];

    const int qt = blockIdx.x;           // 0..124
    const int h  = blockIdx.y;           // 0..15
    const int b  = blockIdx.z;           // 0..1
    const int q0 = qt * 16;

    const int lane = threadIdx.x;
    const int half = lane >> 4;
    const int l16  = lane & 15;
    const int klo  = half * 2;

    const float scale = 0.125f;          // d^-0.5, d=64

    const float* Qb = qh + (size_t)b * SEQ * HID + (size_t)h * DH;
    const float* Kb = kh + (size_t)b * SEQ * HID + (size_t)h * DH;
    const float* Vb = vh + (size_t)b * SEQ * HID + (size_t)h * DH;

    // Preload Q fragments (A layout: row = l16, K split by lane half), scaled.
    v2f qa[16];
#pragma unroll
    for (int j = 0; j < 16; ++j) {
        const int kd = j * 4 + klo;
        const float2 qv = *reinterpret_cast<const float2*>(
            Qb + (size_t)(q0 + l16) * HID + kd);
        qa[j].x = qv.x * scale;
        qa[j].y = qv.y * scale;
    }

    // Per-accumulator-row bias values (row = i + 8*half), both cmask branches.
    float rb0[8], rb1[8];
#pragma unroll
    for (int i = 0; i < 8; ++i) {
        const int r  = q0 + i + 8 * half;
        const int dg = degv[r / 20];
        rb0[i] = tbl[dg * 2 + 0];
        rb1[i] = tbl[dg * 2 + 1];
    }

    v8f o0 = {}, o1 = {}, o2 = {}, o3 = {};
    float mrow[8], lrow[8];
#pragma unroll
    for (int i = 0; i < 8; ++i) { mrow[i] = -1e30f; lrow[i] = 0.f; }

    const int ldr = lane >> 1;            // K/V tile load row (2 lanes/row)
    const int ldc = (lane & 1) * 32;      // column base

    for (int k0 = 0; k0 < SEQ; k0 += 16) {
        // ---- stage K,V tile [16 x 64] into LDS ----
#if USE_ASYNC_LDS
#pragma unroll
        for (int j = 0; j < 8; ++j) {
            const int c = ldc + j * 4;
            const unsigned kd = lds_off(&Ks[ldr][c]);
            const unsigned vd = lds_off(&Vs[ldr][c]);
            const unsigned long long kaddr =
                (unsigned long long)(uintptr_t)(Kb + (size_t)(k0 + ldr) * HID + c);
            const unsigned long long vaddr =
                (unsigned long long)(uintptr_t)(Vb + (size_t)(k0 + ldr) * HID + c);
            asm volatile("global_load_async_to_lds_b128 %0, %1, off"
                         :: "v"(kd), "v"(kaddr) : "memory");
            asm volatile("global_load_async_to_lds_b128 %0, %1, off"
                         :: "v"(vd), "v"(vaddr) : "memory");
        }
        asm volatile("s_wait_asynccnt 0x0" ::: "memory");
#else
#pragma unroll
        for (int j = 0; j < 8; ++j) {
            const int c = ldc + j * 4;
            float4 kv = *reinterpret_cast<const float4*>(
                Kb + (size_t)(k0 + ldr) * HID + c);
            float4 vv = *reinterpret_cast<const float4*>(
                Vb + (size_t)(k0 + ldr) * HID + c);
            *reinterpret_cast<float4*>(&Ks[ldr][c]) = kv;
            *reinterpret_cast<float4*>(&Vs[ldr][c]) = vv;
        }
#endif
        if (k0 + 16 < SEQ) {  // prefetch next tile into cache
            __builtin_prefetch(Kb + (size_t)(k0 + 16 + ldr) * HID + ldc, 0, 3);
            __builtin_prefetch(Vb + (size_t)(k0 + 16 + ldr) * HID + ldc, 0, 3);
        }
        __syncthreads();

        // ---- scores S[16q x 16k] = (Q*scale) @ K^T ----
        v8f s = {};
#pragma unroll
        for (int j = 0; j < 16; ++j) {
            v2f bf;
            bf.x = Ks[l16][j * 4 + klo];      // B[d][k] = K[k][d]
            bf.y = Ks[l16][j * 4 + klo + 1];
            s = WMMA_F32(qa[j], bf, s);
        }

        // ---- add edge bias: table[deg[q/20]][cmask[k/20]] ----
        const int cmv = cmask[(k0 + l16) / 20];
#pragma unroll
        for (int i = 0; i < 8; ++i) s[i] += cmv ? rb1[i] : rb0[i];

        // ---- online softmax (row stats across the 16 lanes of each half) --
#pragma unroll
        for (int i = 0; i < 8; ++i) {
            float mi = s[i];
#pragma unroll
            for (int off = 1; off < 16; off <<= 1)
                mi = fmaxf(mi, __shfl_xor(mi, off, 32));
            const float mnew = fmaxf(mrow[i], mi);
            const float p    = __expf(s[i] - mnew);
            float       sum  = p;
#pragma unroll
            for (int off = 1; off < 16; off <<= 1)
                sum += __shfl_xor(sum, off, 32);
            const float corr = __expf(mrow[i] - mnew);
            lrow[i] = lrow[i] * corr + sum;
            mrow[i] = mnew;
            o0[i] *= corr; o1[i] *= corr; o2[i] *= corr; o3[i] *= corr;
            Ps[i + 8 * half][l16] = p;     // C-layout -> LDS for A re-layout
        }
        __syncthreads();

        // ---- O += P @ V  (A = P[16x16], B = V[16 x 64] in 4 d-chunks) ----
#pragma unroll
        for (int kk = 0; kk < 16; kk += 4) {
            v2f a;
            a.x = Ps[l16][kk + klo];
            a.y = Ps[l16][kk + klo + 1];
            v2f b0, b1, b2, b3;
            b0.x = Vs[kk + klo][ 0 + l16]; b0.y = Vs[kk + klo + 1][ 0 + l16];
            b1.x = Vs[kk + klo][16 + l16]; b1.y = Vs[kk + klo + 1][16 + l16];
            b2.x = Vs[kk + klo][32 + l16]; b2.y = Vs[kk + klo + 1][32 + l16];
            b3.x = Vs[kk + klo][48 + l16]; b3.y = Vs[kk + klo + 1][48 + l16];
            o0 = WMMA_F32(a, b0, o0);
            o1 = WMMA_F32(a, b1, o1);
            o2 = WMMA_F32(a, b2, o2);
            o3 = WMMA_F32(a, b3, o3);
        }
        __syncthreads();
    }

    // ---- normalize and write [16 x 64] back into [B,S,H*64] layout ----
    float* Ob = ao + ((size_t)b * SEQ + q0) * HID + (size_t)h * DH;
#pragma unroll
    for (int i = 0; i < 8; ++i) {
        const float inv = 1.0f / lrow[i];
        const int   r   = i + 8 * half;
        Ob[(size_t)r * HID +  0 + l16] = o0[i] * inv;
        Ob[(size_t)r * HID + 16 + l16] = o1[i] * inv;
        Ob[(size_t)r * HID + 32 + l16] = o2[i] * inv;
        Ob[(size_t)r * HID + 48 + l16] = o3[i] * inv;
    }
}

// ------------------------------ launch -------------------------------------
extern "C" void kernel_launch(void* const* d_in, const int* in_sizes, int n_in,
                              void* d_out, int out_size, void* d_ws, size_t ws_size,
                              hipStream_t stream) {
    const int*   c_mask   = (const int*)  d_in[0];
    const int*   degv     = (const int*)  d_in[1];
    const float* q        = (const float*)d_in[2];
    const float* k        = (const float*)d_in[3];
    const float* v        = (const float*)d_in[4];
    const float* Wq       = (const float*)d_in[5];
    const float* bq       = (const float*)d_in[6];
    const float* Wk       = (const float*)d_in[7];
    const float* bk       = (const float*)d_in[8];
    const float* Wv       = (const float*)d_in[9];
    const float* bv       = (const float*)d_in[10];
    const float* Wo       = (const float*)d_in[11];
    const float* bo       = (const float*)d_in[12];
    const float* edge_emb = (const float*)d_in[13];
    const float* mask_emb = (const float*)d_in[14];

    float* ws  = (float*)d_ws;
    float* qh  = ws;
    float* kh  = ws + SEG;
    float* vh  = ws + 2 * SEG;
    float* ao  = ws + 3 * SEG;
    float* tbl = ws + 4 * SEG;          // 128 floats

    const dim3 gemm_grid(HID / 128, (MROWS + 63) / 64);   // (8, 63)
    const dim3 gemm_blk(256);

    proj_gemm_kernel<<<gemm_grid, gemm_blk, 0, stream>>>(q, Wq, bq, qh, MROWS);
    proj_gemm_kernel<<<gemm_grid, gemm_blk, 0, stream>>>(k, Wk, bk, kh, MROWS);
    proj_gemm_kernel<<<gemm_grid, gemm_blk, 0, stream>>>(v, Wv, bv, vh, MROWS);
    bias_table_kernel<<<64, 256, 0, stream>>>(edge_emb, mask_emb, tbl);

    const dim3 attn_grid(SEQ / 16, NHEAD, NB);    // (125, 16, 2)
    attn_flash_kernel<<<attn_grid, 32, 0, stream>>>(qh, kh, vh, c_mask, degv,
                                                    tbl, ao);

    proj_gemm_kernel<<<gemm_grid, gemm_blk, 0, stream>>>(ao, Wo, bo,
                                                         (float*)d_out, MROWS);
}